// BaseConvHeterogene_65549790872161
// MI455X (gfx1250) — compile-verified
//
#include <hip/hip_runtime.h>

#define N_NODES 50000
#define N_EDGES 800000
#define DIM 128
#define BN_EPS 1e-5f

typedef __attribute__((ext_vector_type(2))) float v2f;
typedef __attribute__((ext_vector_type(8))) float v8f;

// ---------------------------------------------------------------- degrees ---
__global__ __launch_bounds__(256) void k_init_deg(float* __restrict__ deg) {
    int i = blockIdx.x * 256 + threadIdx.x;
    if (i < N_NODES) deg[i] = 1.0f;  // self-loop contributes 1
}

__global__ __launch_bounds__(256) void k_count_deg(const int* __restrict__ ei,
                                                   float* __restrict__ deg) {
    int e = blockIdx.x * 256 + threadIdx.x;
    if (e < N_EDGES) {
        int r = ei[e];  // row = ei[0][e]
        (void)__hip_atomic_fetch_add(&deg[r], 1.0f, __ATOMIC_RELAXED,
                                     __HIP_MEMORY_SCOPE_AGENT);
    }
}

__global__ __launch_bounds__(256) void k_finalize_dinv(float* __restrict__ deg,
                                                       float* __restrict__ stats) {
    int i = blockIdx.x * 256 + threadIdx.x;
    if (i < N_NODES) deg[i] = rsqrtf(deg[i]);           // deg >= 1 always
    if (blockIdx.x == 0) stats[threadIdx.x] = 0.0f;     // zero 256 stat slots
}

// ------------------------------------------------------- xw = x @ W (WMMA) ---
// One wave per 16x16 output tile; K=128 via 32 x V_WMMA_F32_16X16X4_F32.
__global__ __launch_bounds__(256) void k_gemm_wmma(const float* __restrict__ x,
                                                   const float* __restrict__ W,
                                                   float* __restrict__ xw) {
    const int lane = threadIdx.x & 31;
    const int wave = threadIdx.x >> 5;
    const int tile = blockIdx.x * 8 + wave;   // 390 blocks * 8 waves = 3120 tiles
    const int mt   = tile >> 3;               // 390 M-tiles
    const int nt   = tile & 7;                // 8 N-tiles
    const int half = lane >> 4;               // 0: lanes 0-15, 1: lanes 16-31
    const int mn   = lane & 15;

    const float* arow = x + (mt * 16 + mn) * DIM;   // A: row mn, K pair by half
    const float* bcol = W + nt * 16 + mn;           // B: col mn, K pair by half

    v8f c = {};
#pragma unroll
    for (int k = 0; k < DIM; k += 4) {
        const int kk = k + 2 * half;
        v2f a;                    // A 16x4: vgpr0=K(kk), vgpr1=K(kk+1)
        a.x = arow[kk];
        a.y = arow[kk + 1];
        v2f b;                    // B 4x16: mirrored row striping
        b.x = bcol[kk * DIM];
        b.y = bcol[(kk + 1) * DIM];
        c = __builtin_amdgcn_wmma_f32_16x16x4_f32(false, a, false, b,
                                                  (short)0, c, false, false);
    }

    // C/D 16x16 f32: vgpr j -> row j (lanes 0-15) / row j+8 (lanes 16-31)
    float* dst = xw + (mt * 16 + half * 8) * DIM + nt * 16 + mn;
#pragma unroll
    for (int j = 0; j < 8; ++j) dst[j * DIM] = c[j];
}

// --------------------------------- out = bias + self-loop term (init d_out) ---
__global__ __launch_bounds__(256) void k_init_out(const float* __restrict__ xw,
                                                  const float* __restrict__ dinv,
                                                  const float* __restrict__ bias,
                                                  float* __restrict__ out) {
    int idx = blockIdx.x * 256 + threadIdx.x;   // 25000 blocks, N*D exact
    int d = idx & (DIM - 1);
    int i = idx >> 7;
    float di = dinv[i];
    out[idx] = bias[d] + xw[idx] * di * di;
}

// ----------------------------------------------- edge gather + scatter-add ---
// One wave per edge: 32 lanes x float4 = 128 features.
__global__ __launch_bounds__(256) void k_scatter(const int* __restrict__ ei,
                                                 const float* __restrict__ xw,
                                                 const float* __restrict__ dinv,
                                                 float* __restrict__ out) {
    const int lane = threadIdx.x & 31;
    const int wave = threadIdx.x >> 5;
    const int e = blockIdx.x * 8 + wave;        // 100000 blocks * 8 = 800000 exact
    const int r = ei[e];
    const int c = ei[N_EDGES + e];
    const float w = dinv[r] * dinv[c];
    const float4 v = *(const float4*)(xw + c * DIM + lane * 4);
    float* dst = out + r * DIM + lane * 4;
    (void)__hip_atomic_fetch_add(dst + 0, v.x * w, __ATOMIC_RELAXED, __HIP_MEMORY_SCOPE_AGENT);
    (void)__hip_atomic_fetch_add(dst + 1, v.y * w, __ATOMIC_RELAXED, __HIP_MEMORY_SCOPE_AGENT);
    (void)__hip_atomic_fetch_add(dst + 2, v.z * w, __ATOMIC_RELAXED, __HIP_MEMORY_SCOPE_AGENT);
    (void)__hip_atomic_fetch_add(dst + 3, v.w * w, __ATOMIC_RELAXED, __HIP_MEMORY_SCOPE_AGENT);
}

// ------------------------------------------------------- batch-norm stats ---
// 256 threads = 2 rows x 128 features; 250 blocks x 200 rows = 50000 rows.
__global__ __launch_bounds__(256) void k_bn_stats(const float* __restrict__ out,
                                                  float* __restrict__ stats) {
    const int d  = threadIdx.x & 127;
    const int rs = threadIdx.x >> 7;
    const int r0 = blockIdx.x * 200;
    float s = 0.0f, sq = 0.0f;
    for (int r = r0 + rs; r < r0 + 200; r += 2) {
        float v = out[r * DIM + d];
        s += v;
        sq += v * v;
    }
    (void)__hip_atomic_fetch_add(&stats[d],       s,  __ATOMIC_RELAXED, __HIP_MEMORY_SCOPE_AGENT);
    (void)__hip_atomic_fetch_add(&stats[DIM + d], sq, __ATOMIC_RELAXED, __HIP_MEMORY_SCOPE_AGENT);
}

// ------------------------------------------- BN apply + ReLU + residual -----
__global__ __launch_bounds__(256) void k_bn_apply(const float* __restrict__ x,
                                                  const float* __restrict__ gamma,
                                                  const float* __restrict__ beta,
                                                  const float* __restrict__ stats,
                                                  float* __restrict__ out) {
    int idx = blockIdx.x * 256 + threadIdx.x;   // 25000 blocks, N*D exact
    int d = idx & (DIM - 1);
    const float invN = 1.0f / (float)N_NODES;
    float mean = stats[d] * invN;
    float var  = stats[DIM + d] * invN - mean * mean;  // biased, matches jnp.var
    float y = (out[idx] - mean) * rsqrtf(var + BN_EPS) * gamma[d] + beta[d];
    out[idx] = fmaxf(y, 0.0f) + x[idx];
}

// ---------------------------------------------------------------------------
extern "C" void kernel_launch(void* const* d_in, const int* in_sizes, int n_in,
                              void* d_out, int out_size, void* d_ws, size_t ws_size,
                              hipStream_t stream) {
    const float* x     = (const float*)d_in[0];
    const int*   ei    = (const int*)d_in[1];   // [2, E] flattened: row then col
    const float* W     = (const float*)d_in[2];
    const float* bias  = (const float*)d_in[3];
    const float* gamma = (const float*)d_in[4];
    const float* beta  = (const float*)d_in[5];
    float*       out   = (float*)d_out;

    float* xw    = (float*)d_ws;                 // N*D floats (25.6 MB)
    float* dinv  = xw + (size_t)N_NODES * DIM;   // N floats
    float* stats = dinv + N_NODES;               // 256 floats (sum, sumsq)

    k_init_deg     <<<(N_NODES + 255) / 256, 256, 0, stream>>>(dinv);
    k_count_deg    <<<(N_EDGES + 255) / 256, 256, 0, stream>>>(ei, dinv);
    k_finalize_dinv<<<(N_NODES + 255) / 256, 256, 0, stream>>>(dinv, stats);
    k_gemm_wmma    <<<(N_NODES / 16) / 8 * 8 / 8, 256, 0, stream>>>(x, W, xw); // 390
    k_init_out     <<<(N_NODES * DIM) / 256, 256, 0, stream>>>(xw, dinv, bias, out);
    k_scatter      <<<N_EDGES / 8, 256, 0, stream>>>(ei, xw, dinv, out);
    k_bn_stats     <<<N_NODES / 200, 256, 0, stream>>>(out, stats);
    k_bn_apply     <<<(N_NODES * DIM) / 256, 256, 0, stream>>>(x, gamma, beta, stats, out);
}